// EfficientGlobalPointer_40209483825374
// MI455X (gfx1250) — compile-verified
//
#include <hip/hip_runtime.h>
#include <math.h>

#define HIDDEN 1024
#define NHEADS 12
#define HEAD   64
#define SEQ    1024
#define BATCH  4
#define ROWS   (BATCH * SEQ)   // 4096
#define PDIM   (2 * HEAD)      // 128
#define QDIM   (2 * NHEADS)    // 24

typedef __attribute__((ext_vector_type(2))) float v2f;
typedef __attribute__((ext_vector_type(8))) float v8f;

// D = A(16x4 f32) * B(4x16 f32) + C(16x16 f32), wave32 CDNA5 WMMA.
__device__ __forceinline__ v8f wmma_f32(v2f a, v2f b, v8f c) {
  return __builtin_amdgcn_wmma_f32_16x16x4_f32(
      /*neg_a=*/false, a, /*neg_b=*/false, b,
      /*c_mod=*/(short)0, c, /*reuse_a=*/false, /*reuse_b=*/false);
}

// ---------------------------------------------------------------------------
// Stage 1: seq = inputs @ Wp + bp          (M=4096, K=1024, N=128)
// One block = 16 output rows x 128 cols; wave w owns the 16-col tile w.
// A-fragment (ISA 7.12.2, 32-bit A 16x4): lanes 0-15 -> M=lane, K=0/1 in
// v0/v1; lanes 16-31 -> K=2/3.  B mirrors it (K in lane-half+vgpr, N=lane).
// ---------------------------------------------------------------------------
__global__ void egp_proj_kernel(const float* __restrict__ in,
                                const float* __restrict__ Wp,
                                const float* __restrict__ bp,
                                float* __restrict__ seq) {
  const int lane = threadIdx.x & 31;
  const int wave = threadIdx.x >> 5;    // 0..7 -> N tile
  const int half = lane >> 4;           // lane-half selects K pair
  const int l15  = lane & 15;
  const int m_base = blockIdx.x * 16;
  const int n_base = wave * 16;
  const int nn = n_base + l15;

  const float* arow = in + (size_t)(m_base + l15) * HIDDEN;

  v8f acc = {};
  for (int kk = 0; kk < HIDDEN / 4; ++kk) {
    const int k0 = kk * 4 + half * 2;
    v2f a = *(const v2f*)(arow + k0);               // A[m][k0], A[m][k0+1]
    v2f b;
    b.x = Wp[(size_t)k0 * PDIM + nn];               // B[k0][n]
    b.y = Wp[(size_t)(k0 + 1) * PDIM + nn];         // B[k0+1][n]
    acc = wmma_f32(a, b, acc);
  }

  const float bias = bp[nn];
#pragma unroll
  for (int r = 0; r < 8; ++r) {                     // C/D: vgpr r -> M=r / r+8
    const int m = m_base + r + 8 * half;
    seq[(size_t)m * PDIM + nn] = acc[r] + bias;
  }
}

// ---------------------------------------------------------------------------
// Stage 2: per row -- RoPE the q/k halves, and project 128 -> 24 bias terms.
// Row staged in LDS; 64 threads (2 waves) per row.
// ---------------------------------------------------------------------------
__global__ void egp_rope_bias_kernel(const float* __restrict__ seq,
                                     const float* __restrict__ Wq,
                                     const float* __restrict__ bq,
                                     float* __restrict__ qr,
                                     float* __restrict__ kr,
                                     float* __restrict__ qbias,
                                     float* __restrict__ kbias) {
  const int row = blockIdx.x;             // 0..4095 = b*1024 + pos
  const int pos = row & (SEQ - 1);
  const int t   = threadIdx.x;            // 0..63

  __shared__ float srow[PDIM];
  srow[t]      = seq[(size_t)row * PDIM + t];
  srow[t + 64] = seq[(size_t)row * PDIM + t + 64];
  __syncthreads();

  // RoPE: t<32 handles q pair i=t, t>=32 handles k pair i=t-32.
  const int  isK  = t >> 5;
  const int  i    = t & 31;
  const int  base = isK * HEAD;
  const float s0 = srow[base + 2 * i];
  const float s1 = srow[base + 2 * i + 1];
  // div_i = 10000^(-2i/64)
  const float inv = expf(-(float)(2 * i) * (logf(10000.0f) / (float)HEAD));
  const float ang = (float)pos * inv;
  const float c = cosf(ang), s = sinf(ang);
  float* dst = isK ? kr : qr;
  dst[(size_t)row * HEAD + 2 * i]     = s0 * c - s1 * s;
  dst[(size_t)row * HEAD + 2 * i + 1] = s1 * c + s0 * s;

  // bias_input[row][j] = seq[row] . Wq[:,j] + bq[j], then /2 and split
  if (t < QDIM) {
    float sum = bq[t];
#pragma unroll 8
    for (int k = 0; k < PDIM; ++k) sum += srow[k] * Wq[(size_t)k * QDIM + t];
    sum *= 0.5f;
    if ((t & 1) == 0) qbias[(size_t)row * NHEADS + (t >> 1)] = sum;
    else              kbias[(size_t)row * NHEADS + (t >> 1)] = sum;
  }
}

// ---------------------------------------------------------------------------
// Stage 3: logits[b,h,m,n] = (q[b,m,:].k[b,n,:])/8 + qb[b,m,h] + kb[b,n,h]
//                            (mask -> -inf) - (m>n)*1e12
// One wave per 16x16 (m,n) tile; K=64 via 16 chained f32 WMMAs. The tile is
// then transposed through LDS so each lane owns a 4-wide n-quad of a row,
// letting the 12 head planes stream out as global_store_b128 (each store
// instruction covers 8 rows x 64B instead of 32 scattered dwords).
// ---------------------------------------------------------------------------
__global__ void egp_logits_kernel(const float* __restrict__ qr,
                                  const float* __restrict__ kr,
                                  const float* __restrict__ qbias,
                                  const float* __restrict__ kbias,
                                  const int* __restrict__ mask,
                                  float* __restrict__ out) {
  const int lane = threadIdx.x & 31;
  const int wave = threadIdx.x >> 5;
  const int half = lane >> 4;
  const int l15  = lane & 15;

  const int batch  = blockIdx.z;
  const int n_base = blockIdx.x * 16;
  const int m_base = (blockIdx.y * 8 + wave) * 16;

  __shared__ float tile[8][16][16];   // 1KB per wave

  const float* aq = qr + ((size_t)batch * SEQ + m_base + l15) * HEAD;
  const float* bk = kr + ((size_t)batch * SEQ + n_base + l15) * HEAD;

  v8f acc = {};
#pragma unroll
  for (int kk = 0; kk < HEAD / 4; ++kk) {
    const int k0 = kk * 4 + half * 2;
    v2f a = *(const v2f*)(aq + k0);   // Q[m][k] fragment
    v2f b = *(const v2f*)(bk + k0);   // B[k][n] = K[n][k] fragment
    acc = wmma_f32(a, b, acc);
  }

  // Stage scaled tile to LDS: lane holds (m = r + 8*half, n = l15) in acc[r].
#pragma unroll
  for (int r = 0; r < 8; ++r)
    tile[wave][r + 8 * half][l15] = acc[r] * 0.125f;  // 1/sqrt(64)
  __syncthreads();

  // Transposed ownership: lane -> rows (lane>>2) and (lane>>2)+8,
  // n-quad starting at (lane&3)*4.
  const int qm = lane >> 2;
  const int n0 = (lane & 3) * 4;
  const int n  = n_base + n0;

  // Per-lane key-side bias quad for all heads (rows are 48B -> 16B aligned).
  float kbq[4][NHEADS];
  int   mq[4];
#pragma unroll
  for (int j = 0; j < 4; ++j) {
    const float4* kp = (const float4*)(kbias + (size_t)(batch * SEQ + n + j) * NHEADS);
    const float4 k0v = kp[0], k1v = kp[1], k2v = kp[2];
    kbq[j][0] = k0v.x; kbq[j][1]  = k0v.y; kbq[j][2]  = k0v.z; kbq[j][3]  = k0v.w;
    kbq[j][4] = k1v.x; kbq[j][5]  = k1v.y; kbq[j][6]  = k1v.z; kbq[j][7]  = k1v.w;
    kbq[j][8] = k2v.x; kbq[j][9]  = k2v.y; kbq[j][10] = k2v.z; kbq[j][11] = k2v.w;
    mq[j] = mask[batch * SEQ + n + j];
  }

#pragma unroll
  for (int round = 0; round < 2; ++round) {
    const int m = m_base + qm + round * 8;
    const int mask_m = mask[batch * SEQ + m];
    const float4 v = *(const float4*)&tile[wave][qm + round * 8][n0];  // ds_load_b128

    const float4* qp = (const float4*)(qbias + (size_t)(batch * SEQ + m) * NHEADS);
    const float4 q0v = qp[0], q1v = qp[1], q2v = qp[2];
    float qbv[NHEADS] = {q0v.x, q0v.y, q0v.z, q0v.w,
                         q1v.x, q1v.y, q1v.z, q1v.w,
                         q2v.x, q2v.y, q2v.z, q2v.w};

    float cz[4], dead[4];
#pragma unroll
    for (int j = 0; j < 4; ++j) {
      cz[j]   = (m > n + j) ? 1.0e12f : 0.0f;
      dead[j] = (mask_m == 0) || (mq[j] == 0) ? 1.0f : 0.0f;
    }

#pragma unroll
    for (int h = 0; h < NHEADS; ++h) {
      float4 o;
      o.x = v.x + qbv[h] + kbq[0][h];
      o.y = v.y + qbv[h] + kbq[1][h];
      o.z = v.z + qbv[h] + kbq[2][h];
      o.w = v.w + qbv[h] + kbq[3][h];
      if (dead[0] != 0.0f) o.x = -INFINITY;
      if (dead[1] != 0.0f) o.y = -INFINITY;
      if (dead[2] != 0.0f) o.z = -INFINITY;
      if (dead[3] != 0.0f) o.w = -INFINITY;
      o.x -= cz[0]; o.y -= cz[1]; o.z -= cz[2]; o.w -= cz[3];  // -inf stays -inf
      float* dst = out + (((size_t)(batch * NHEADS + h)) * SEQ + m) * SEQ + n;
      *(float4*)dst = o;   // global_store_b128
    }
  }
}

extern "C" void kernel_launch(void* const* d_in, const int* in_sizes, int n_in,
                              void* d_out, int out_size, void* d_ws, size_t ws_size,
                              hipStream_t stream) {
  const float* inputs = (const float*)d_in[0];   // (4,1024,1024) f32
  const int*   mask   = (const int*)  d_in[1];   // (4,1024) i32
  const float* Wp     = (const float*)d_in[2];   // (1024,128)
  const float* bp     = (const float*)d_in[3];   // (128,)
  const float* Wq     = (const float*)d_in[4];   // (128,24)
  const float* bq     = (const float*)d_in[5];   // (24,)
  float* out = (float*)d_out;                    // (4,12,1024,1024) f32

  char* ws = (char*)d_ws;
  float* seq = (float*)ws;  ws += (size_t)ROWS * PDIM * sizeof(float);   // 2 MB
  float* qr  = (float*)ws;  ws += (size_t)ROWS * HEAD * sizeof(float);   // 1 MB
  float* kr  = (float*)ws;  ws += (size_t)ROWS * HEAD * sizeof(float);   // 1 MB
  float* qb  = (float*)ws;  ws += (size_t)ROWS * NHEADS * sizeof(float); // 192 KB
  float* kb  = (float*)ws;

  egp_proj_kernel<<<dim3(ROWS / 16), dim3(256), 0, stream>>>(inputs, Wp, bp, seq);
  egp_rope_bias_kernel<<<dim3(ROWS), dim3(64), 0, stream>>>(seq, Wq, bq, qr, kr, qb, kb);
  egp_logits_kernel<<<dim3(SEQ / 16, SEQ / 128, BATCH), dim3(256), 0, stream>>>(
      qr, kr, qb, kb, mask, out);
}